// ManualGATv2Layer_87608742903867
// MI455X (gfx1250) — compile-verified
//
#include <hip/hip_runtime.h>
#include <hip/hip_bf16.h>

#define IN_DIM    128
#define HOUT      64      // H*D
#define NHEAD     4
#define NEG_SLOPE 0.2f

typedef float v2f __attribute__((ext_vector_type(2)));
typedef float v8f __attribute__((ext_vector_type(8)));

// ---------------------------------------------------------------------------
// Zero out[N*64], amax bits [N*4], asum [N*4]
// ---------------------------------------------------------------------------
__global__ void gat_zero(float* __restrict__ out, int* __restrict__ amax_bits,
                         float* __restrict__ asum, int n_out, int n_nh) {
  int i = blockIdx.x * blockDim.x + threadIdx.x;
  if (i < n_out) out[i] = 0.0f;
  if (i < n_nh)  { amax_bits[i] = 0; asum[i] = 0.0f; }
}

// ---------------------------------------------------------------------------
// Projection: hs = x @ Wsrc^T, ht = x @ Wtgt^T via V_WMMA_F32_16X16X4_F32.
// One wave handles a 16-node x 16-outcol tile for BOTH weights.
//   A (16x4 f32): lane<16 -> row=lane, K={k0,k0+1}; lane>=16 -> K={k0+2,k0+3}
//   B (4x16 f32): lane<16 -> col=lane, K={k0,k0+1}; lane>=16 -> K={k0+2,k0+3}
//   D: VGPR v -> row = v + 8*(lane>>4), col = lane&15
// ---------------------------------------------------------------------------
__global__ __launch_bounds__(256) void gat_proj_wmma(
    const float* __restrict__ x, const float* __restrict__ Wsrc,
    const float* __restrict__ Wtgt, float* __restrict__ hs,
    float* __restrict__ ht, int ntile) {
  const int lane  = threadIdx.x & 31;
  const int wave  = (blockIdx.x * blockDim.x + threadIdx.x) >> 5;
  const int tile  = wave >> 2;
  if (tile >= ntile) return;             // wave-uniform exit: EXEC stays all-1s
  const int ocol  = (wave & 3) << 4;     // 0,16,32,48
  const int row   = lane & 15;
  const int khalf = lane >> 4;           // which K pair this half-wave owns

  const float* xrow  = x    + (size_t)(tile * 16 + row) * IN_DIM + khalf * 2;
  const float* wsrow = Wsrc + (size_t)(ocol + row)      * IN_DIM + khalf * 2;
  const float* wtrow = Wtgt + (size_t)(ocol + row)      * IN_DIM + khalf * 2;

  v8f acc_s = {0.f,0.f,0.f,0.f,0.f,0.f,0.f,0.f};
  v8f acc_t = {0.f,0.f,0.f,0.f,0.f,0.f,0.f,0.f};

#pragma unroll 4
  for (int k0 = 0; k0 < IN_DIM; k0 += 4) {
    v2f a  = *(const v2f*)(xrow  + k0);
    v2f bs = *(const v2f*)(wsrow + k0);
    v2f bt = *(const v2f*)(wtrow + k0);
    acc_s = __builtin_amdgcn_wmma_f32_16x16x4_f32(
        false, a, false, bs, (short)0, acc_s, false, false);
    acc_t = __builtin_amdgcn_wmma_f32_16x16x4_f32(
        false, a, false, bt, (short)0, acc_t, false, false);
  }

  float* ds = hs + (size_t)(tile * 16 + khalf * 8) * HOUT + ocol + row;
  float* dt = ht + (size_t)(tile * 16 + khalf * 8) * HOUT + ocol + row;
#pragma unroll
  for (int v = 0; v < 8; ++v) {
    ds[(size_t)v * HOUT] = acc_s[v];
    dt[(size_t)v * HOUT] = acc_t[v];
  }
}

// ---------------------------------------------------------------------------
// Per-edge: msg = leaky(hs[src]+ht[tgt]); alpha[e,h] = <msg, attn[h]>;
// segment max into amax via int-bit atomicMax (amax init 0 == bits 0; IEEE
// non-negative floats order correctly as ints; negatives are negative ints
// and never beat 0 -> matches reference max-with-zero semantics).
// ---------------------------------------------------------------------------
__global__ void gat_edge_alpha(const int* __restrict__ ei,
                               const float* __restrict__ hs,
                               const float* __restrict__ ht,
                               const float* __restrict__ attn,
                               float* __restrict__ alpha,
                               int* __restrict__ amax_bits, int E) {
  int e = blockIdx.x * blockDim.x + threadIdx.x;
  if (e >= E) return;
  int s = ei[e];
  int t = ei[E + e];
  const float4* ps = (const float4*)(hs + (size_t)s * HOUT);
  const float4* pt = (const float4*)(ht + (size_t)t * HOUT);
  const float4* pa = (const float4*)attn;
#pragma unroll
  for (int h = 0; h < NHEAD; ++h) {
    float acc = 0.0f;
#pragma unroll
    for (int q = 0; q < 4; ++q) {
      float4 vs = ps[h * 4 + q];
      float4 vt = pt[h * 4 + q];
      float4 va = pa[h * 4 + q];
      float m;
      m = vs.x + vt.x; m = (m >= 0.f) ? m : NEG_SLOPE * m; acc += m * va.x;
      m = vs.y + vt.y; m = (m >= 0.f) ? m : NEG_SLOPE * m; acc += m * va.y;
      m = vs.z + vt.z; m = (m >= 0.f) ? m : NEG_SLOPE * m; acc += m * va.z;
      m = vs.w + vt.w; m = (m >= 0.f) ? m : NEG_SLOPE * m; acc += m * va.w;
    }
    alpha[(size_t)e * NHEAD + h] = acc;
    atomicMax(&amax_bits[t * NHEAD + h], __float_as_int(acc));
  }
}

// ---------------------------------------------------------------------------
// alpha = exp(alpha - amax[tgt]); asum[tgt] += alpha
// ---------------------------------------------------------------------------
__global__ void gat_edge_exp(const int* __restrict__ ei,
                             const int* __restrict__ amax_bits,
                             float* __restrict__ alpha,
                             float* __restrict__ asum, int E) {
  int e = blockIdx.x * blockDim.x + threadIdx.x;
  if (e >= E) return;
  int t = ei[E + e];
#pragma unroll
  for (int h = 0; h < NHEAD; ++h) {
    float m  = __int_as_float(amax_bits[t * NHEAD + h]);
    float a  = alpha[(size_t)e * NHEAD + h];
    float ex = expf(a - m);
    alpha[(size_t)e * NHEAD + h] = ex;
    atomicAdd(&asum[t * NHEAD + h], ex);
  }
}

// ---------------------------------------------------------------------------
// out[tgt] += hs[src] * (alpha / (asum[tgt] + 1e-8))
// ---------------------------------------------------------------------------
__global__ void gat_edge_scatter(const int* __restrict__ ei,
                                 const float* __restrict__ hs,
                                 const float* __restrict__ alpha,
                                 const float* __restrict__ asum,
                                 float* __restrict__ out, int E) {
  int e = blockIdx.x * blockDim.x + threadIdx.x;
  if (e >= E) return;
  int s = ei[e];
  int t = ei[E + e];
  const float4* ps = (const float4*)(hs + (size_t)s * HOUT);
  float w[NHEAD];
#pragma unroll
  for (int h = 0; h < NHEAD; ++h)
    w[h] = alpha[(size_t)e * NHEAD + h] / (asum[t * NHEAD + h] + 1e-8f);
  float* ob = out + (size_t)t * HOUT;
#pragma unroll
  for (int h = 0; h < NHEAD; ++h) {
#pragma unroll
    for (int q = 0; q < 4; ++q) {
      float4 v = ps[h * 4 + q];
      int base = h * 16 + q * 4;
      atomicAdd(ob + base + 0, v.x * w[h]);
      atomicAdd(ob + base + 1, v.y * w[h]);
      atomicAdd(ob + base + 2, v.z * w[h]);
      atomicAdd(ob + base + 3, v.w * w[h]);
    }
  }
}

// ---------------------------------------------------------------------------
extern "C" void kernel_launch(void* const* d_in, const int* in_sizes, int n_in,
                              void* d_out, int out_size, void* d_ws, size_t ws_size,
                              hipStream_t stream) {
  const float* x    = (const float*)d_in[0];
  const int*   ei   = (const int*)  d_in[1];   // [2,E] (src row, tgt row)
  const float* Wsrc = (const float*)d_in[2];   // [64,128]
  const float* Wtgt = (const float*)d_in[3];   // [64,128]
  const float* attn = (const float*)d_in[4];   // [4,16]

  const int N = in_sizes[0] / IN_DIM;
  const int E = in_sizes[1] / 2;
  float* out = (float*)d_out;

  // workspace layout (fp32 elements): hs | ht | alpha | amax(int) | asum
  float* hs    = (float*)d_ws;
  float* ht    = hs + (size_t)N * HOUT;
  float* alpha = ht + (size_t)N * HOUT;
  int*   amax  = (int*)(alpha + (size_t)E * NHEAD);
  float* asum  = (float*)(amax + (size_t)N * NHEAD);

  // 1) zero accumulators + output
  int n_out = N * HOUT, n_nh = N * NHEAD;
  gat_zero<<<(n_out + 255) / 256, 256, 0, stream>>>(out, amax, asum, n_out, n_nh);

  // 2) WMMA projections (one wave per 16x16 tile, 4 col-tiles per node-tile)
  int ntile = (N + 15) / 16;
  long long threads = (long long)ntile * 4 * 32;
  int blocks = (int)((threads + 255) / 256);
  gat_proj_wmma<<<blocks, 256, 0, stream>>>(x, Wsrc, Wtgt, hs, ht, ntile);

  // 3..5) edge phase
  int eblocks = (E + 255) / 256;
  gat_edge_alpha  <<<eblocks, 256, 0, stream>>>(ei, hs, ht, attn, alpha, amax, E);
  gat_edge_exp    <<<eblocks, 256, 0, stream>>>(ei, amax, alpha, asum, E);
  gat_edge_scatter<<<eblocks, 256, 0, stream>>>(ei, hs, alpha, asum, out, E);
}